// GNN_36086315221296
// MI455X (gfx1250) — compile-verified
//
#include <hip/hip_runtime.h>

#define D 128
#define WAVES_PER_BLOCK 8
#define SCATTER_BLOCK (WAVES_PER_BLOCK * 32)
#define DEPTH 4  // async pipeline depth per wave (power of two)

// Exact vector type the builtin expects (GCC vector_size style, per diagnostic)
typedef int v4i __attribute__((vector_size(16)));
typedef __attribute__((address_space(1))) v4i* global_v4i_p;
typedef __attribute__((address_space(3))) v4i* lds_v4i_p;

// ---------------------------------------------------------------------------
// Async global -> LDS copy of 16B per lane (512B per wave = one x-row).
// CDNA5 path: GLOBAL_LOAD_ASYNC_TO_LDS_B128, tracked by ASYNCcnt.
// ---------------------------------------------------------------------------
__device__ __forceinline__ void async_gather_16B(const float* gsrc, float* lds_dst) {
#if __has_builtin(__builtin_amdgcn_global_load_async_to_lds_b128)
    __builtin_amdgcn_global_load_async_to_lds_b128(
        (global_v4i_p)gsrc,   // C-style cast: const-cast + bitcast + addrspacecast
        (lds_v4i_p)lds_dst,
        0, 0);
#else
    // Generic LDS pointer: low 32 bits are the LDS byte offset.
    unsigned lds_off = (unsigned)(size_t)lds_dst;
    asm volatile("global_load_async_to_lds_b128 %0, %1, off"
                 :: "v"(lds_off), "v"(gsrc)
                 : "memory");
#endif
}

__device__ __forceinline__ void wait_async_le3() {
    // immediate-only SOPP; keep <= DEPTH-1 outstanding (oldest load retired)
    asm volatile("s_wait_asynccnt 3" ::: "memory");
}

// ---------------------------------------------------------------------------
// Scatter kernel: one wave per edge (grid-stride). Pipeline DEPTH gathers of
// x[row[e]] (512B each) through a private LDS slot, then issue 4 non-returning
// f32 atomics per lane into out[col[e], 0:128]. Lane 0 also counts the edge
// into out[col[e], 128] (reclaimed by the finalize pass).
// ---------------------------------------------------------------------------
__global__ __launch_bounds__(SCATTER_BLOCK)
void gnn_scatter(const float* __restrict__ x, const int* __restrict__ es,
                 float* __restrict__ out, int E) {
    __shared__ float sh[WAVES_PER_BLOCK * DEPTH * D];  // 16 KB

    const int lane   = threadIdx.x & 31;
    const int wib    = threadIdx.x >> 5;
    const int gwave  = blockIdx.x * WAVES_PER_BLOCK + wib;
    const int nwaves = gridDim.x * WAVES_PER_BLOCK;
    if (gwave >= E) return;

    float* slots = sh + wib * (DEPTH * D);
    const int* __restrict__ rowp = es + E;  // es[1] = row

    // ---- prologue: fill the pipeline (clamp tail so DEPTH stay in flight) --
    int e_issue = gwave;
#pragma unroll
    for (int s = 0; s < DEPTH; ++s) {
        int ec = (e_issue < E) ? e_issue : (E - 1);
        int r  = rowp[__builtin_amdgcn_readfirstlane(ec)];   // s_load (uniform)
        async_gather_16B(x + (size_t)r * D + lane * 4,
                         slots + s * D + lane * 4);
        e_issue += nwaves;
    }

    // ---- steady state ------------------------------------------------------
    int slot = 0;
    for (int e = gwave; e < E; e += nwaves) {
        wait_async_le3();  // oldest gather has landed in LDS

        const int c = es[__builtin_amdgcn_readfirstlane(e)];  // col (uniform)
        const float4 v =
            *reinterpret_cast<const float4*>(slots + slot * D + lane * 4);

        float* dst = out + (size_t)c * (2 * D) + lane * 4;
        unsafeAtomicAdd(dst + 0, v.x);
        unsafeAtomicAdd(dst + 1, v.y);
        unsafeAtomicAdd(dst + 2, v.z);
        unsafeAtomicAdd(dst + 3, v.w);
        if (lane == 0)
            unsafeAtomicAdd(out + (size_t)c * (2 * D) + D, 1.0f);

        // keep the consume (ds_load + atomics) ahead of the slot refill
        asm volatile("" ::: "memory");

        int ec = (e_issue < E) ? e_issue : (E - 1);
        int r  = rowp[__builtin_amdgcn_readfirstlane(ec)];
        async_gather_16B(x + (size_t)r * D + lane * 4,
                         slots + slot * D + lane * 4);
        e_issue += nwaves;
        slot = (slot + 1) & (DEPTH - 1);
    }
    // leftover async ops drained by implicit wait-idle at s_endpgm
}

// ---------------------------------------------------------------------------
// Finalize: one wave per node. Reads the count stashed at out[n,128], scales
// the accumulated sum, and writes the second half (= x[n] if count>0 else 0).
// The count cell is read before lane 0 overwrites it (data dependence forces
// the load-before-store order within the wave).
// ---------------------------------------------------------------------------
__global__ __launch_bounds__(256)
void gnn_finalize(const float* __restrict__ x, float* __restrict__ out, int N) {
    const int lane = threadIdx.x & 31;
    const int wib  = threadIdx.x >> 5;
    const int n    = blockIdx.x * WAVES_PER_BLOCK + wib;
    if (n >= N) return;

    float* o = out + (size_t)n * (2 * D);
    const float c   = o[D];                       // edge count (as float)
    const float inv = 1.0f / fmaxf(c, 1.0f);
    const bool  has = (c > 0.0f);

    float4 s  = *reinterpret_cast<const float4*>(o + lane * 4);
    float4 xv = *reinterpret_cast<const float4*>(x + (size_t)n * D + lane * 4);

    s.x *= inv; s.y *= inv; s.z *= inv; s.w *= inv;
    float4 sec = has ? xv : make_float4(0.f, 0.f, 0.f, 0.f);

    *reinterpret_cast<float4*>(o + lane * 4)     = s;
    *reinterpret_cast<float4*>(o + D + lane * 4) = sec;
}

// ---------------------------------------------------------------------------
extern "C" void kernel_launch(void* const* d_in, const int* in_sizes, int n_in,
                              void* d_out, int out_size, void* d_ws, size_t ws_size,
                              hipStream_t stream) {
    const float* x  = (const float*)d_in[0];
    const int*   es = (const int*)d_in[1];     // [2, E] flat: col then row
    float*       out = (float*)d_out;          // [N, 256]

    const int N = in_sizes[0] / D;
    const int E = in_sizes[1] / 2;

    // zero accumulators (graph-capture-safe)
    (void)hipMemsetAsync(out, 0, (size_t)out_size * sizeof(float), stream);

    // scatter: 1024 blocks x 8 waves -> 8192 waves, ~98 edges each
    gnn_scatter<<<1024, SCATTER_BLOCK, 0, stream>>>(x, es, out, E);

    // finalize: one wave per node
    const int fblocks = (N + WAVES_PER_BLOCK - 1) / WAVES_PER_BLOCK;
    gnn_finalize<<<fblocks, 256, 0, stream>>>(x, out, N);
}